// SGConvRegression_80418967650355
// MI455X (gfx1250) — compile-verified
//
#include <hip/hip_runtime.h>
#include <hip/hip_bf16.h>

typedef __attribute__((ext_vector_type(2))) float v2f;
typedef __attribute__((ext_vector_type(8))) float v8f;

#define TPB 256

// ---- init degree (self loop => 1), csr fill cursors, batch accumulators ----
__global__ void init_kernel(int* __restrict__ deg, int* __restrict__ fill,
                            float* __restrict__ sums, float* __restrict__ cnts,
                            int n, int g) {
  int i = blockIdx.x * blockDim.x + threadIdx.x;
  if (i < n) { deg[i] = 1; fill[i] = 0; }
  if (i < g) { sums[i] = 0.0f; cnts[i] = 0.0f; }
}

// ---- in-degree count over col ----
__global__ void count_kernel(const int* __restrict__ col, int* __restrict__ deg, int e) {
  int i = blockIdx.x * blockDim.x + threadIdx.x;
  if (i < e) atomicAdd(&deg[col[i]], 1);
}

// ---- dis = rsqrt(deg) (deg >= 1 always due to self loop) ----
__global__ void dis_kernel(const int* __restrict__ deg, float* __restrict__ dis, int n) {
  int i = blockIdx.x * blockDim.x + threadIdx.x;
  if (i < n) dis[i] = rsqrtf((float)deg[i]);
}

// ---- single-block exclusive scan of (deg-1) -> CSR row pointers ----
__global__ void scan_kernel(const int* __restrict__ deg, int* __restrict__ ptr, int n) {
  __shared__ int sh[1024];
  __shared__ int carry;
  if (threadIdx.x == 0) carry = 0;
  __syncthreads();
  for (int base = 0; base < n; base += 1024) {
    int i = base + (int)threadIdx.x;
    int v = (i < n) ? (deg[i] - 1) : 0;
    sh[threadIdx.x] = v;
    __syncthreads();
    for (int off = 1; off < 1024; off <<= 1) {
      int t = (threadIdx.x >= (unsigned)off) ? sh[threadIdx.x - off] : 0;
      __syncthreads();
      sh[threadIdx.x] += t;
      __syncthreads();
    }
    if (i < n) ptr[i] = carry + sh[threadIdx.x] - v;   // exclusive
    __syncthreads();
    if (threadIdx.x == 0) carry += sh[1023];
    __syncthreads();
  }
  if (threadIdx.x == 0) ptr[n] = carry;                // == E
}

// ---- scatter edges into col-sorted CSR, precompute edge weights ----
__global__ void fill_kernel(const int* __restrict__ row, const int* __restrict__ col,
                            const float* __restrict__ dis, const int* __restrict__ ptr,
                            int* __restrict__ fill, int* __restrict__ crow,
                            float* __restrict__ cw, int e) {
  int i = blockIdx.x * blockDim.x + threadIdx.x;
  if (i < e) {
    int c = col[i], r = row[i];
    int p = ptr[c] + atomicAdd(&fill[c], 1);
    crow[p] = r;
    cw[p] = dis[r] * dis[c];
  }
}

// ---- one hop: one wave32 per destination node; lane owns 4 of 128 features.
//      Atomic-free: register accumulate over the node's in-edge list. ----
__global__ void hop_kernel(const float* __restrict__ xin, float* __restrict__ xout,
                           const float* __restrict__ dis, const int* __restrict__ ptr,
                           const int* __restrict__ crow, const float* __restrict__ cw,
                           int n) {
  int wave = (blockIdx.x * blockDim.x + threadIdx.x) >> 5;   // wave32
  int lane = threadIdx.x & 31;
  if (wave >= n) return;
  int node = wave;
  const float4* xin4 = (const float4*)xin;                   // F=128 -> 32 float4/node
  float ds = dis[node];
  float sw = ds * ds;                                        // self-loop weight
  float4 sv = xin4[(size_t)node * 32 + lane];
  float4 acc;
  acc.x = sw * sv.x; acc.y = sw * sv.y; acc.z = sw * sv.z; acc.w = sw * sv.w;
  int beg = ptr[node], end = ptr[node + 1];
  for (int j = beg; j < end; ++j) {
    int r = crow[j];
    float wgt = cw[j];
    float4 v = xin4[(size_t)r * 32 + lane];
    acc.x += wgt * v.x; acc.y += wgt * v.y; acc.z += wgt * v.z; acc.w += wgt * v.w;
  }
  ((float4*)xout)[(size_t)node * 32 + lane] = acc;
}

// ---- dense projection via fp32 WMMA (16x16x4) + scatter into batch bins.
//      One wave per 16-node tile. A layout (32-bit, 16x4): lanes 0-15 hold
//      K=0,1 in v[0],v[1]; lanes 16-31 hold K=2,3. B broadcasts W[k..k+3]
//      across all 16 N columns, so every output column equals the dot. ----
__global__ void project_kernel(const float* __restrict__ x, const float* __restrict__ W,
                               const float* __restrict__ b, const int* __restrict__ batch,
                               float* __restrict__ sums, float* __restrict__ cnts, int n) {
  int wave = (blockIdx.x * blockDim.x + threadIdx.x) >> 5;
  int lane = threadIdx.x & 31;
  int base = wave * 16;
  if (base >= n) return;                 // whole-wave uniform exit (EXEC stays all-1 in body)
  int mrow = lane & 15;
  int khalf = (lane >> 4) << 1;          // 0 for lanes 0-15, 2 for lanes 16-31
  int m = base + mrow;
  if (m > n - 1) m = n - 1;              // clamp tail reads (N=100000 is 16-aligned anyway)
  v8f c = {};
  for (int k = 0; k < 128; k += 4) {
    const float* xr = x + (size_t)m * 128 + k + khalf;
    v2f a, bb;
    a.x = xr[0];           a.y = xr[1];
    bb.x = W[k + khalf];   bb.y = W[k + khalf + 1];
    c = __builtin_amdgcn_wmma_f32_16x16x4_f32(
        /*neg_a=*/false, a, /*neg_b=*/false, bb,
        /*c_mod=*/(short)0, c, /*reuse_a=*/false, /*reuse_b=*/false);
  }
  float bias = b[0];
  // C/D layout: lane 0 holds C[0..7, 0] in c[0..7]; lane 16 holds C[8..15, 0].
  if (lane == 0 || lane == 16) {
    int m0 = (lane == 0) ? 0 : 8;
    for (int r = 0; r < 8; ++r) {
      int node = base + m0 + r;
      if (node < n) {
        float val = c[r] + bias;
        int g = batch[node];
        atomicAdd(&sums[g], val);
        atomicAdd(&cnts[g], 1.0f);
      }
    }
  }
}

// ---- final mean per graph ----
__global__ void finalize_kernel(const float* __restrict__ sums, const float* __restrict__ cnts,
                                float* __restrict__ out, int g) {
  int i = blockIdx.x * blockDim.x + threadIdx.x;
  if (i < g) out[i] = sums[i] / fmaxf(cnts[i], 1.0f);
}

extern "C" void kernel_launch(void* const* d_in, const int* in_sizes, int n_in,
                              void* d_out, int out_size, void* d_ws, size_t ws_size,
                              hipStream_t stream) {
  const float* x     = (const float*)d_in[0];   // [N, F]
  const float* W     = (const float*)d_in[1];   // [F, 1]
  const float* b     = (const float*)d_in[2];   // [1]
  const int*   ei    = (const int*)d_in[3];     // [2, E]
  const int*   batch = (const int*)d_in[4];     // [N]

  const int F = in_sizes[1];        // 128
  const int N = in_sizes[4];
  const int E = in_sizes[3] / 2;
  const int G = out_size;
  const int* row = ei;
  const int* col = ei + E;
  (void)F; (void)n_in; (void)ws_size;

  // carve workspace (~117 MB)
  char* w = (char*)d_ws;
  float* xA  = (float*)w; w += (size_t)N * 128 * sizeof(float);
  float* xB  = (float*)w; w += (size_t)N * 128 * sizeof(float);
  float* dis = (float*)w; w += (size_t)N * sizeof(float);
  int*   deg = (int*)w;   w += (size_t)N * sizeof(int);
  int*   ptr = (int*)w;   w += (size_t)(N + 1) * sizeof(int);
  int*   fil = (int*)w;   w += (size_t)N * sizeof(int);
  int*   crw = (int*)w;   w += (size_t)E * sizeof(int);
  float* cwt = (float*)w; w += (size_t)E * sizeof(float);
  float* sums = (float*)w; w += (size_t)G * sizeof(float);
  float* cnts = (float*)w;

  int nb = (N + TPB - 1) / TPB;
  int eb = (E + TPB - 1) / TPB;

  init_kernel<<<nb, TPB, 0, stream>>>(deg, fil, sums, cnts, N, G);
  count_kernel<<<eb, TPB, 0, stream>>>(col, deg, E);
  dis_kernel<<<nb, TPB, 0, stream>>>(deg, dis, N);
  scan_kernel<<<1, 1024, 0, stream>>>(deg, ptr, N);
  fill_kernel<<<eb, TPB, 0, stream>>>(row, col, dis, ptr, fil, crw, cwt, E);

  int hopBlocks = (N + 7) / 8;                  // 8 waves/block, 1 node/wave
  hop_kernel<<<hopBlocks, TPB, 0, stream>>>(x,  xA, dis, ptr, crw, cwt, N);
  hop_kernel<<<hopBlocks, TPB, 0, stream>>>(xA, xB, dis, ptr, crw, cwt, N);
  hop_kernel<<<hopBlocks, TPB, 0, stream>>>(xB, xA, dis, ptr, crw, cwt, N);

  int tiles = (N + 15) / 16;                    // one wave per 16-node tile
  project_kernel<<<(tiles * 32 + TPB - 1) / TPB, TPB, 0, stream>>>(xA, W, b, batch, sums, cnts, N);
  finalize_kernel<<<(G + TPB - 1) / TPB, TPB, 0, stream>>>(sums, cnts, (float*)d_out, G);
}